// ForwardFunctionalMap_83880711291319
// MI455X (gfx1250) — compile-verified
//
#include <hip/hip_runtime.h>

typedef __attribute__((ext_vector_type(2))) float v2f;
typedef __attribute__((ext_vector_type(8))) float v8f;
typedef int b128vec __attribute__((vector_size(16)));   // matches builtin's V4i

#define B_   16
#define K_   128
#define N_   8000
#define D_   256
#define LMBDA_ 1000.0f
#define MP_  132   /* augmented-matrix pitch: 128 cols + rhs col + align pad */

#define AS1 __attribute__((address_space(1)))
#define AS3 __attribute__((address_space(3)))

// ---- CDNA5 async global->LDS copy (ASYNCcnt-tracked, no VGPR staging) ------
#if defined(__has_builtin)
#  if __has_builtin(__builtin_amdgcn_global_load_async_to_lds_b128)
#    define HAVE_ASYNC_COPY 1
#  else
#    define HAVE_ASYNC_COPY 0
#  endif
#else
#  define HAVE_ASYNC_COPY 0
#endif

__device__ __forceinline__ void async_copy16(const float* g, float* l) {
#if HAVE_ASYNC_COPY
  __builtin_amdgcn_global_load_async_to_lds_b128(
      (AS1 b128vec*)g, (AS3 b128vec*)l, 0, 0);
#else
  *(float4*)l = *(const float4*)g;
#endif
}

__device__ __forceinline__ void async_join() {
#if HAVE_ASYNC_COPY
  asm volatile("s_wait_asynccnt 0x0" ::: "memory");
#endif
}

// ---------------------------------------------------------------------------
// Kernel 1: spectral projection  sdescr[b,d,k] = sum_n pinv[b,k,n]*descr[b,d,n]
// NT-GEMM over contiguous N. f32 WMMA 16x16x4.
// grid (B, (D/64)*(K/64), 2) = 256 WGs; 128 threads = 4 waves;
// each wave computes 4 16x16 tiles of a 64(d) x 64(k) output block.
// ---------------------------------------------------------------------------
__global__ __launch_bounds__(128) void project_kernel(
    const float* __restrict__ pinv_a, const float* __restrict__ descr_a,
    const float* __restrict__ pinv_b, const float* __restrict__ descr_b,
    float* __restrict__ sdescr_a, float* __restrict__ sdescr_b)
{
  const int b      = blockIdx.x;
  const int dBase  = (blockIdx.y >> 1) * 64;   // 0,64,128,192
  const int kStrip = (blockIdx.y & 1) * 64;    // 0 or 64
  const float* __restrict__ pinv  = blockIdx.z ? pinv_b   : pinv_a;
  const float* __restrict__ descr = blockIdx.z ? descr_b  : descr_a;
  float*       __restrict__ outp  = blockIdx.z ? sdescr_b : sdescr_a;

  __shared__ float sA[64][68];    // descr tile [d][n], padded pitch (272B rows)
  __shared__ float sB[64][68];    // pinv  tile [k][n]

  const int tid   = threadIdx.x;
  const int lane  = tid & 31;
  const int wave  = tid >> 5;     // 0..3
  const int m     = lane & 15;
  const int hi    = lane >> 4;    // 0: lanes 0-15, 1: lanes 16-31
  const int dTile = wave;         // 16-row slab of the 64-row strip

  const float* gD = descr + (size_t)b * D_ * N_ + (size_t)dBase  * N_;
  const float* gP = pinv  + (size_t)b * K_ * N_ + (size_t)kStrip * N_;

  v8f acc[4] = {};

  for (int n0 = 0; n0 < N_; n0 += 64) {
    __syncthreads();
#pragma unroll
    for (int p = 0; p < 8; ++p) {           // 64x64 descr chunk (1024 float4)
      int idx = p * 128 + tid;
      int row = idx >> 4;
      int c4  = (idx & 15) * 4;
      async_copy16(gD + (size_t)row * N_ + n0 + c4, &sA[row][c4]);
    }
#pragma unroll
    for (int p = 0; p < 8; ++p) {           // 64x64 pinv chunk
      int idx = p * 128 + tid;
      int row = idx >> 4;
      int c4  = (idx & 15) * 4;
      async_copy16(gP + (size_t)row * N_ + n0 + c4, &sB[row][c4]);
    }
    if (n0 + 64 < N_) {                     // stream-ahead -> global_prefetch_b8
      __builtin_prefetch(gD + (size_t)(tid & 63) * N_ + n0 + 64, 0, 0);
      __builtin_prefetch(gP + (size_t)(tid & 63) * N_ + n0 + 64, 0, 0);
    }
    async_join();
    __syncthreads();
#pragma unroll 4
    for (int nn = 0; nn < 64; nn += 4) {
      // A 16x4 frag: lanes 0-15 hold K=0/1, lanes 16-31 hold K=2/3
      v2f av;
      av.x = sA[dTile * 16 + m][nn + 2 * hi];
      av.y = sA[dTile * 16 + m][nn + 2 * hi + 1];
#pragma unroll
      for (int t = 0; t < 4; ++t) {
        // B 4x16 frag: v0 holds rows K=hi, v1 holds rows K=hi+2 (col = m)
        v2f bv;
        bv.x = sB[t * 16 + m][nn + hi];
        bv.y = sB[t * 16 + m][nn + hi + 2];
        acc[t] = __builtin_amdgcn_wmma_f32_16x16x4_f32(
            false, av, false, bv, (short)0, acc[t], false, false);
      }
    }
  }

  // C/D layout: VGPR r -> M=r (lanes 0-15) / M=8+r (lanes 16-31), N=lane&15
  float* gO = outp + (size_t)b * D_ * K_;
#pragma unroll
  for (int t = 0; t < 4; ++t) {
    int kcol = kStrip + t * 16 + m;
#pragma unroll
    for (int r = 0; r < 8; ++r) {
      int drow = dBase + dTile * 16 + r + 8 * hi;
      gO[(size_t)drow * K_ + kcol] = acc[t][r];
    }
  }
}

// ---------------------------------------------------------------------------
// Kernel 2: Gram matrices  AA[pair][b][k][l] = sum_d X[b,d,k] * Y[b,d,l]
// pair 0: (a,a)  1: (b,a)  2: (b,b)  3: (a,b).  Contraction over D=256.
// grid (4, B); one WG computes the full 128x128; wave w owns row-tile w.
// ---------------------------------------------------------------------------
__global__ __launch_bounds__(256) void gram_kernel(
    const float* __restrict__ sa, const float* __restrict__ sb,
    float* __restrict__ AA)
{
  const int pair = blockIdx.x;
  const int b    = blockIdx.y;
  const float* X = (pair == 1 || pair == 2) ? sb : sa;
  const float* Y = (pair == 0 || pair == 1) ? sa : sb;
  float* outp = AA + ((size_t)pair * B_ + b) * K_ * K_;

  __shared__ float sX[32][132];
  __shared__ float sY[32][132];

  const int tid  = threadIdx.x;
  const int lane = tid & 31;
  const int wave = tid >> 5;
  const int m    = lane & 15;
  const int hi   = lane >> 4;
  const int kTile = wave;

  const float* gX = X + (size_t)b * D_ * K_;
  const float* gY = Y + (size_t)b * D_ * K_;

  v8f acc[8] = {};

  for (int d0 = 0; d0 < D_; d0 += 32) {
    __syncthreads();
#pragma unroll
    for (int p = 0; p < 4; ++p) {           // 32x128 chunks of both operands
      int idx = p * 256 + tid;
      int row = idx >> 5;
      int c4  = (idx & 31) * 4;
      async_copy16(gX + (size_t)(d0 + row) * K_ + c4, &sX[row][c4]);
      async_copy16(gY + (size_t)(d0 + row) * K_ + c4, &sY[row][c4]);
    }
    async_join();
    __syncthreads();
#pragma unroll
    for (int dd = 0; dd < 32; dd += 4) {
      v2f av;                                // A[m][c] = X[d0+dd+c][kTile*16+m]
      av.x = sX[dd + 2 * hi][kTile * 16 + m];
      av.y = sX[dd + 2 * hi + 1][kTile * 16 + m];
#pragma unroll
      for (int lt = 0; lt < 8; ++lt) {
        v2f bv;                              // B[c][l] = Y[d0+dd+c][lt*16+m]
        bv.x = sY[dd + hi][lt * 16 + m];
        bv.y = sY[dd + hi + 2][lt * 16 + m];
        acc[lt] = __builtin_amdgcn_wmma_f32_16x16x4_f32(
            false, av, false, bv, (short)0, acc[lt], false, false);
      }
    }
  }

#pragma unroll
  for (int lt = 0; lt < 8; ++lt) {
    int lcol = lt * 16 + m;
#pragma unroll
    for (int r = 0; r < 8; ++r) {
      int krow = kTile * 16 + r + 8 * hi;
      outp[(size_t)krow * K_ + lcol] = acc[lt][r];
    }
  }
}

// ---------------------------------------------------------------------------
// Kernel 3: resolvent masks.  mask_ab[b,k,l] = f(gb=eb[k]/s, ga=ea[l]/s),
// mask_ba[b,k,l] = f(gb=ea[k]/s, ga=eb[l]/s), s = max(ea,eb). GAMMA==1.
// ---------------------------------------------------------------------------
__global__ __launch_bounds__(128) void mask_kernel(
    const float* __restrict__ evals_a, const float* __restrict__ evals_b,
    float* __restrict__ mask_ab, float* __restrict__ mask_ba)
{
  const int b = blockIdx.x, t = threadIdx.x;
  __shared__ float ea[128], eb[128], red[128];
  ea[t]  = evals_a[b * K_ + t];
  eb[t]  = evals_b[b * K_ + t];
  red[t] = fmaxf(ea[t], eb[t]);
  __syncthreads();
  for (int s = 64; s > 0; s >>= 1) {
    if (t < s) red[t] = fmaxf(red[t], red[t + s]);
    __syncthreads();
  }
  const float inv_scale = 1.0f / red[0];

  float* mab = mask_ab + (size_t)b * K_ * K_;
  float* mba = mask_ba + (size_t)b * K_ * K_;

  const float gaA = ea[t] * inv_scale;            // column term, mask_ab
  const float dA  = 1.0f / (gaA * gaA + 1.0f);
  const float cA  = gaA * dA;
  const float gaB = eb[t] * inv_scale;            // column term, mask_ba
  const float dB  = 1.0f / (gaB * gaB + 1.0f);
  const float cB  = gaB * dB;

  for (int k = 0; k < K_; ++k) {
    float gb = eb[k] * inv_scale;                 // row term, mask_ab
    float db = 1.0f / (gb * gb + 1.0f);
    float re = gb * db - cA, im = db - dA;
    mab[k * K_ + t] = re * re + im * im;

    float gb2 = ea[k] * inv_scale;                // row term, mask_ba
    float db2 = 1.0f / (gb2 * gb2 + 1.0f);
    float re2 = gb2 * db2 - cB, im2 = db2 - dB;
    mba[k * K_ + t] = re2 * re2 + im2 * im2;
  }
}

// ---------------------------------------------------------------------------
// Kernel 4: per-(dir,b,row) 128x128 solve, pivot-free Gauss-Jordan on the
// augmented system in dynamic LDS (128 x 132 floats = 67,584 B < 320 KB/WG).
// A is SPD (Gram + nonneg diag) so no pivoting is required.
// grid (K, B, 2); 256 threads = 8 waves, wave g owns rows g, g+8, ...
// ---------------------------------------------------------------------------
__global__ __launch_bounds__(256) void solve_kernel(
    const float* __restrict__ AA,        // [4][B][K][K]: 0=aa 1=ba 2=bb 3=ab
    const float* __restrict__ mask_ab, const float* __restrict__ mask_ba,
    float* __restrict__ outp)            // [2][B][K][K]
{
  extern __shared__ float M[];           // pitch MP_=132, col 128 = rhs
  const int i = blockIdx.x, b = blockIdx.y, dir = blockIdx.z;

  const float* gA = AA + ((size_t)(dir ? 2 : 0) * B_ + b) * K_ * K_;
  const float* gR = AA + ((size_t)(dir ? 3 : 1) * B_ + b) * K_ * K_ + (size_t)i * K_;
  const float* gM = (dir ? mask_ba : mask_ab) + (size_t)b * K_ * K_ + (size_t)i * K_;
  float* gOut = outp + ((size_t)dir * B_ + b) * K_ * K_ + (size_t)i * K_;

  const int tid = threadIdx.x;
  for (int p = 0; p < 16; ++p) {         // 128x128 matrix: 4096 float4 chunks
    int idx4 = p * 256 + tid;
    int r = idx4 >> 5;                   // 32 float4 per row
    int c = (idx4 & 31) * 4;
    async_copy16(gA + (size_t)r * K_ + c, &M[r * MP_ + c]);
  }
  if (tid < 128) M[tid * MP_ + 128] = gR[tid];     // rhs column
  async_join();
  __syncthreads();
  if (tid < 128) M[tid * MP_ + tid] += LMBDA_ * gM[tid];
  __syncthreads();

  const int lane = tid & 31, grp = tid >> 5;
  for (int k = 0; k < 128; ++k) {
    const float invp = 1.0f / M[k * MP_ + k];
    const float* rowk = &M[k * MP_];
    for (int r = grp; r < 128; r += 8) {
      if (r == k) continue;
      float f = M[r * MP_ + k] * invp;
      for (int c = k + 1 + lane; c <= 128; c += 32)
        M[r * MP_ + c] -= f * rowk[c];
    }
    __syncthreads();                      // after Jordan sweep: A is diagonal
  }
  if (tid < 128)
    gOut[tid] = M[tid * MP_ + 128] / M[tid * MP_ + tid];
}

// ---------------------------------------------------------------------------
extern "C" void kernel_launch(void* const* d_in, const int* in_sizes, int n_in,
                              void* d_out, int out_size, void* d_ws, size_t ws_size,
                              hipStream_t stream)
{
  const float* evals_a = (const float*)d_in[0];
  const float* pinv_a  = (const float*)d_in[1];
  const float* descr_a = (const float*)d_in[2];
  const float* evals_b = (const float*)d_in[3];
  const float* pinv_b  = (const float*)d_in[4];
  const float* descr_b = (const float*)d_in[5];
  float* outp = (float*)d_out;
  float* ws   = (float*)d_ws;

  // workspace layout (floats): total 2,621,440 (10.5 MB)
  float* sdescr_a = ws;                                  // B*D*K
  float* sdescr_b = sdescr_a + (size_t)B_ * D_ * K_;     // B*D*K
  float* AA       = sdescr_b + (size_t)B_ * D_ * K_;     // 4*B*K*K
  float* mask_ab  = AA + (size_t)4 * B_ * K_ * K_;       // B*K*K
  float* mask_ba  = mask_ab + (size_t)B_ * K_ * K_;      // B*K*K

  project_kernel<<<dim3(B_, 8, 2), 128, 0, stream>>>(
      pinv_a, descr_a, pinv_b, descr_b, sdescr_a, sdescr_b);
  gram_kernel<<<dim3(4, B_), 256, 0, stream>>>(sdescr_a, sdescr_b, AA);
  mask_kernel<<<dim3(B_), 128, 0, stream>>>(evals_a, evals_b, mask_ab, mask_ba);
  solve_kernel<<<dim3(K_, B_, 2), 256, (size_t)(128 * MP_) * sizeof(float), stream>>>(
      AA, mask_ab, mask_ba, outp);
}